// NodeGCN_1589137899686
// MI455X (gfx1250) — compile-verified
//
#include <hip/hip_runtime.h>

#define NN 100000
#define EE 3200000
#define FIN 512
#define HH 20
#define CC 10
#define SP 32   // padded row stride for h/agg/o: 128B = one cacheline per row

typedef __attribute__((ext_vector_type(2))) float v2f;
typedef __attribute__((ext_vector_type(8))) float v8f;

// ---------------- degree / normalization coefficients ----------------

__global__ void __launch_bounds__(256) k_init_deg(float* __restrict__ deg) {
  int i = blockIdx.x * 256 + threadIdx.x;
  if (i < NN) deg[i] = 1.0f;  // self loop contributes 1
}

__global__ void __launch_bounds__(256) k_count_deg(const int* __restrict__ dst,
                                                   float* __restrict__ deg) {
  int e = blockIdx.x * 256 + threadIdx.x;
  if (e < EE) atomicAdd(&deg[dst[e]], 1.0f);
}

__global__ void __launch_bounds__(256) k_deg_to_dis(float* __restrict__ deg) {
  int i = blockIdx.x * 256 + threadIdx.x;
  if (i < NN) {
    float d = deg[i];
    deg[i] = (d > 0.0f) ? rsqrtf(d) : 0.0f;
  }
}

// ---------------- B-fragment packing for V_WMMA_F32_16X16X4_F32 ----------------
// B is 4x16 per step: lane l holds column n = l&15, K slots koff..koff+1, koff = 2*(l>>4).
// Fragment frag = ks*ntiles + tile, laid out as float2[32] per fragment (zero padded).

__global__ void __launch_bounds__(256) k_pack_b(const float* __restrict__ W,
                                                float* __restrict__ P,
                                                int K, int Ncols, int ntiles) {
  int t = blockIdx.x * 256 + threadIdx.x;
  int total = (K / 4) * ntiles * 32;
  if (t >= total) return;
  int lane = t & 31;
  int frag = t >> 5;
  int ks   = frag / ntiles;
  int tile = frag - ks * ntiles;
  int koff = (lane >> 4) * 2;
  int n    = tile * 16 + (lane & 15);
  int kk   = ks * 4 + koff;
  float v0 = (n < Ncols) ? W[kk * Ncols + n] : 0.0f;
  float v1 = (n < Ncols) ? W[(kk + 1) * Ncols + n] : 0.0f;
  P[(size_t)t * 2 + 0] = v0;
  P[(size_t)t * 2 + 1] = v1;
}

// ---------------- GEMM: h = x @ W1 (M=100000, K=512, N=20 -> 2 col tiles) ----------------
// Fuses self-loop init: agg[i][j] = h[i][j] * dis[i]^2. Branch-free epilogue (stride SP).

__global__ void __launch_bounds__(256) k_gemm_x(const float* __restrict__ x,
                                                const float* __restrict__ Bp,
                                                const float* __restrict__ dis,
                                                float* __restrict__ h,
                                                float* __restrict__ agg) {
  int wave = (blockIdx.x * blockDim.x + threadIdx.x) >> 5;
  int lane = threadIdx.x & 31;
  if (wave >= NN / 16) return;          // wave-uniform: EXEC all-1 for WMMA
  int half = lane >> 4;
  int l16  = lane & 15;
  int koff = half * 2;
  int m    = wave * 16 + l16;
  int n    = l16;
  v8f c0 = {}; v8f c1 = {};
  const float* xr = x + (size_t)m * FIN + koff;         // 8B aligned
  const v2f* P = (const v2f*)Bp;
  #pragma unroll 4
  for (int ks = 0; ks < FIN / 4; ++ks) {
    // x is streamed exactly once: non-temporal so reused buffers stay hot in L2
    v2f a  = __builtin_nontemporal_load((const v2f*)(xr + ks * 4));
    v2f b0 = P[(ks * 2 + 0) * 32 + lane];
    v2f b1 = P[(ks * 2 + 1) * 32 + lane];
    c0 = __builtin_amdgcn_wmma_f32_16x16x4_f32(false, a, false, b0, (short)0, c0, false, false);
    c1 = __builtin_amdgcn_wmma_f32_16x16x4_f32(false, a, false, b1, (short)0, c1, false, false);
  }
  int rbase = wave * 16 + half * 8;     // D row = v + 8*(lane>=16)
  #pragma unroll
  for (int v = 0; v < 8; ++v) {
    int row = rbase + v;
    float s = dis[row]; s = s * s;
    size_t r = (size_t)row * SP;
    float h0 = c0[v], h1 = c1[v];
    h[r + n]        = h0;
    h[r + n + 16]   = h1;               // cols 20..31 are pad, never read
    agg[r + n]      = h0 * s;
    agg[r + n + 16] = h1 * s;
  }
}

// ---------------- GEMM: h = o @ W (K=20, N=20), fused self-loop init ----------------

__global__ void __launch_bounds__(256) k_gemm_h(const float* __restrict__ o,
                                                const float* __restrict__ Bp,
                                                const float* __restrict__ dis,
                                                float* __restrict__ h,
                                                float* __restrict__ agg) {
  int wave = (blockIdx.x * blockDim.x + threadIdx.x) >> 5;
  int lane = threadIdx.x & 31;
  if (wave >= NN / 16) return;
  int half = lane >> 4;
  int l16  = lane & 15;
  int koff = half * 2;
  int m    = wave * 16 + l16;
  int n    = l16;
  v8f c0 = {}; v8f c1 = {};
  const float* orow = o + (size_t)m * SP + koff;        // 8B aligned
  const v2f* P = (const v2f*)Bp;
  #pragma unroll
  for (int ks = 0; ks < HH / 4; ++ks) {                 // 5 steps
    v2f a  = *(const v2f*)(orow + ks * 4);
    v2f b0 = P[(ks * 2 + 0) * 32 + lane];
    v2f b1 = P[(ks * 2 + 1) * 32 + lane];
    c0 = __builtin_amdgcn_wmma_f32_16x16x4_f32(false, a, false, b0, (short)0, c0, false, false);
    c1 = __builtin_amdgcn_wmma_f32_16x16x4_f32(false, a, false, b1, (short)0, c1, false, false);
  }
  int rbase = wave * 16 + half * 8;
  #pragma unroll
  for (int v = 0; v < 8; ++v) {
    int row = rbase + v;
    float s = dis[row]; s = s * s;
    size_t r = (size_t)row * SP;
    float h0 = c0[v], h1 = c1[v];
    h[r + n]        = h0;
    h[r + n + 16]   = h1;
    agg[r + n]      = h0 * s;
    agg[r + n + 16] = h1 * s;
  }
}

// ---------------- edge scatter: agg[dst] += h[src] * dis[src]*dis[dst] ----------------
// Rows are cacheline aligned (SP*4 = 128B): each gather touches exactly one line.

__global__ void __launch_bounds__(256) k_agg_edges(const float* __restrict__ h,
                                                   const float* __restrict__ dis,
                                                   const int* __restrict__ src,
                                                   const int* __restrict__ dst,
                                                   float* __restrict__ agg) {
  int e = blockIdx.x * 256 + threadIdx.x;
  if (e >= EE) return;
  int s = src[e];
  int d = dst[e];
  float nrm = dis[s] * dis[d];
  const float4* hr = (const float4*)(h + (size_t)s * SP);
  float* ar = agg + (size_t)d * SP;
  #pragma unroll
  for (int q = 0; q < HH / 4; ++q) {
    float4 v = hr[q];
    atomicAdd(&ar[q * 4 + 0], v.x * nrm);
    atomicAdd(&ar[q * 4 + 1], v.y * nrm);
    atomicAdd(&ar[q * 4 + 2], v.z * nrm);
    atomicAdd(&ar[q * 4 + 3], v.w * nrm);
  }
}

// ---------------- o = relu( (agg + b) / max(||agg+b||_2, eps) ) ----------------

__global__ void __launch_bounds__(256) k_norm_relu(const float* __restrict__ agg,
                                                   const float* __restrict__ bias,
                                                   float* __restrict__ o) {
  int i = blockIdx.x * 256 + threadIdx.x;
  if (i >= NN) return;
  float v[HH];
  float ss = 0.0f;
  const float* ar = agg + (size_t)i * SP;
  #pragma unroll
  for (int j = 0; j < HH; ++j) {
    float t = ar[j] + bias[j];
    v[j] = t;
    ss += t * t;
  }
  float inv = 1.0f / fmaxf(sqrtf(ss), 1e-12f);
  float* orow = o + (size_t)i * SP;
  #pragma unroll
  for (int j = 0; j < HH; ++j) orow[j] = fmaxf(v[j] * inv, 0.0f);
}

// ---------------- head: out = [o1|o2|o3] @ Wl + bl (K=60, N=10 -> 1 col tile) ----------------

__global__ void __launch_bounds__(256) k_final(const float* __restrict__ o1,
                                               const float* __restrict__ o2,
                                               const float* __restrict__ o3,
                                               const float* __restrict__ Bp,
                                               const float* __restrict__ bl,
                                               float* __restrict__ out) {
  int wave = (blockIdx.x * blockDim.x + threadIdx.x) >> 5;
  int lane = threadIdx.x & 31;
  if (wave >= NN / 16) return;
  int half = lane >> 4;
  int l16  = lane & 15;
  int koff = half * 2;
  int m    = wave * 16 + l16;
  int n    = l16;
  v8f c = {};
  const v2f* P = (const v2f*)Bp;
  #pragma unroll
  for (int ks = 0; ks < 15; ++ks) {     // 15 steps; each stays within one layer (20%4==0)
    int layer = ks / 5;
    int kloc  = (ks - layer * 5) * 4;
    const float* ob = (layer == 0) ? o1 : (layer == 1) ? o2 : o3;
    v2f a = *(const v2f*)(ob + (size_t)m * SP + kloc + koff);
    v2f b = P[ks * 32 + lane];          // zero padded for n >= 10
    c = __builtin_amdgcn_wmma_f32_16x16x4_f32(false, a, false, b, (short)0, c, false, false);
  }
  if (n < CC) {
    float bv = bl[n];
    int rbase = wave * 16 + half * 8;
    #pragma unroll
    for (int v = 0; v < 8; ++v)
      out[(size_t)(rbase + v) * CC + n] = c[v] + bv;
  }
}

// ---------------- launcher ----------------

extern "C" void kernel_launch(void* const* d_in, const int* in_sizes, int n_in,
                              void* d_out, int out_size, void* d_ws, size_t ws_size,
                              hipStream_t stream) {
  (void)in_sizes; (void)n_in; (void)out_size; (void)ws_size;
  const float* x  = (const float*)d_in[0];
  const int*   ei = (const int*)d_in[1];
  const float* W1 = (const float*)d_in[2];
  const float* b1 = (const float*)d_in[3];
  const float* W2 = (const float*)d_in[4];
  const float* b2 = (const float*)d_in[5];
  const float* W3 = (const float*)d_in[6];
  const float* b3 = (const float*)d_in[7];
  const float* Wl = (const float*)d_in[8];
  const float* bl = (const float*)d_in[9];
  const int* srcI = ei;        // edge_index[0]
  const int* dstI = ei + EE;   // edge_index[1]
  float* out = (float*)d_out;

  // ws layout (floats): dis[N] | h | agg | o1 | o2 | o3 (each N*SP) | Bp1 | Bp2 | Bp3 | Bpl
  float* ws   = (float*)d_ws;
  float* dis  = ws;                      // 400000B: keeps hbuf 128B aligned
  float* hbuf = dis + NN;
  float* agg  = hbuf + (size_t)NN * SP;
  float* o1   = agg + (size_t)NN * SP;
  float* o2   = o1 + (size_t)NN * SP;
  float* o3   = o2 + (size_t)NN * SP;
  float* Bp1  = o3 + (size_t)NN * SP;            // (512/4)*2*32*2 = 16384 floats
  float* Bp2  = Bp1 + (size_t)(FIN / 4) * 2 * 64;
  float* Bp3  = Bp2 + (size_t)(HH / 4) * 2 * 64; // 640 floats each
  float* Bpl  = Bp3 + (size_t)(HH / 4) * 2 * 64; // 15*64 = 960 floats

  dim3 B(256);
  int gN  = (NN + 255) / 256;
  int gE  = (EE + 255) / 256;
  int gW  = ((NN / 16) * 32 + 255) / 256;  // one wave per 16-row tile
  int gP1 = ((FIN / 4) * 2 * 32 + 255) / 256;
  int gPh = ((HH / 4) * 2 * 32 + 255) / 256;
  int gPl = (15 * 32 + 255) / 256;

  // pack WMMA B fragments (zero padded)
  k_pack_b<<<gP1, B, 0, stream>>>(W1, Bp1, FIN, HH, 2);
  k_pack_b<<<gPh, B, 0, stream>>>(W2, Bp2, HH, HH, 2);
  k_pack_b<<<gPh, B, 0, stream>>>(W3, Bp3, HH, HH, 2);
  k_pack_b<<<gPl, B, 0, stream>>>(Wl, Bpl, 3 * HH, CC, 1);

  // degrees -> dis
  k_init_deg<<<gN, B, 0, stream>>>(dis);
  k_count_deg<<<gE, B, 0, stream>>>(dstI, dis);
  k_deg_to_dis<<<gN, B, 0, stream>>>(dis);

  // layer 1
  k_gemm_x<<<gW, B, 0, stream>>>(x, Bp1, dis, hbuf, agg);
  k_agg_edges<<<gE, B, 0, stream>>>(hbuf, dis, srcI, dstI, agg);
  k_norm_relu<<<gN, B, 0, stream>>>(agg, b1, o1);

  // layer 2
  k_gemm_h<<<gW, B, 0, stream>>>(o1, Bp2, dis, hbuf, agg);
  k_agg_edges<<<gE, B, 0, stream>>>(hbuf, dis, srcI, dstI, agg);
  k_norm_relu<<<gN, B, 0, stream>>>(agg, b2, o2);

  // layer 3
  k_gemm_h<<<gW, B, 0, stream>>>(o2, Bp3, dis, hbuf, agg);
  k_agg_edges<<<gE, B, 0, stream>>>(hbuf, dis, srcI, dstI, agg);
  k_norm_relu<<<gN, B, 0, stream>>>(agg, b3, o3);

  // head
  k_final<<<gW, B, 0, stream>>>(o1, o2, o3, Bpl, bl, out);
}